// GuidedFilterBase_58488864637026
// MI455X (gfx1250) — compile-verified
//
#include <hip/hip_runtime.h>

// Radius-4 box SUM over (8,3,1024,1024) f32, separable, zero-padded edges
// (zero-pad == reference's truncated cumsum-diff windows).
//
// Formulated as banded-matrix GEMMs on V_WMMA_F32_16X16X4_F32:
//   Out(16x16) = R(16x24) * X(24x24) * W(24x16),  R/W = 0/1 band matrices.
// Memory-bound: ~325MB total traffic @ 23.3TB/s ~= 14us floor; 18 WMMAs/tile
// of compute is free. One wave per 16x16 tile, LDS-staged 24x24 halo tile.
//
// v2: band fragments (identical for horizontal-B and vertical-A, and shared
// by both row blocks) are computed ONCE into band[6] instead of per-WMMA,
// halving the v_cmp/v_cndmask VALU setup seen in round 1's histogram.

typedef float v2f __attribute__((ext_vector_type(2)));
typedef float v8f __attribute__((ext_vector_type(8)));

#define RAD   4
#define TILE  16
#define HALO  24          // TILE + 2*RAD
#define IMW   1024
#define IMH   1024
#define XS    25          // LDS row stride for X tile (conflict-free: gcd(25,64)=1)
#define XAREA (HALO * XS) // 600 floats
#define MS    16          // LDS row stride for intermediate M (24x16)
#define WLDS  992         // per-wave LDS floats (600 + 384, padded)
#define WAVES 4           // waves per block

__global__ __launch_bounds__(WAVES * 32)
void box9_wmma(const float* __restrict__ x, float* __restrict__ out) {
  __shared__ float smem[WAVES * WLDS];
  const int lane = threadIdx.x & 31;
  const int wave = threadIdx.x >> 5;
  float* lx = &smem[wave * WLDS]; // 24x25 padded input tile
  float* lm = lx + XAREA;         // 24x16 horizontal-pass intermediate

  const int tile = blockIdx.x * WAVES + wave;
  const int img  = tile >> 12;    // 4096 tiles per 1024x1024 image
  const int rem  = tile & 4095;
  const int ty   = rem >> 6;
  const int tx   = rem & 63;
  const long base = (long)img * (IMW * IMH);
  const int r0 = ty * TILE, c0 = tx * TILE;

  const int n  = lane & 15; // col (B/C/D) or row (A) index held by this lane
  const int kh = lane >> 4; // lane-half selects K pair in A/B fragments

  // ---- Band fragments, computed once. Fragment c covers K = 4c..4c+3.
  // In both A (16x4) and assumed-symmetric B (4x16) layouts, this lane's
  // two VGPRs hold K = 4c + 2*kh and 4c + 2*kh + 1, paired with row/col n.
  // Band value: 1 iff n <= K <= n + 2*RAD.
  v2f band[HALO / 4];
  #pragma unroll
  for (int c = 0; c < HALO / 4; ++c) {
    const int k0 = 4 * c + 2 * kh;
    band[c].x = (k0     >= n && k0     <= n + 2 * RAD) ? 1.0f : 0.0f;
    band[c].y = (k0 + 1 >= n && k0 + 1 <= n + 2 * RAD) ? 1.0f : 0.0f;
  }

  // ---- Stage 24x24 halo tile into LDS, zero-padded at image borders ----
  for (int idx = lane; idx < HALO * HALO; idx += 32) {
    const int lr = idx / HALO, lc = idx - lr * HALO;
    const int gr = r0 - RAD + lr, gc = c0 - RAD + lc;
    float v = 0.0f;
    if (gr >= 0 && gr < IMH && gc >= 0 && gc < IMW)
      v = x[base + (long)gr * IMW + gc];
    lx[lr * XS + lc] = v;
  }
  __syncthreads();

  // ---- Horizontal pass: M[0:24][0:16] = X[0:24][0:24] * W(24x16) ----
  // Two 16-row blocks (rows 0-15 in d0, rows 8-23 in d1), shared B = band.
  v8f d0 = {}; v8f d1 = {};
  #pragma unroll
  for (int c = 0; c < HALO / 4; ++c) {
    const int k0 = 4 * c + 2 * kh;                    // K of first VGPR
    v2f a0, a1;                                       // A: row n of block, K=k0,k0+1
    a0.x = lx[n * XS + k0];        a0.y = lx[n * XS + k0 + 1];
    a1.x = lx[(8 + n) * XS + k0];  a1.y = lx[(8 + n) * XS + k0 + 1];
    d0 = __builtin_amdgcn_wmma_f32_16x16x4_f32(false, a0, false, band[c], (short)0, d0, false, false);
    d1 = __builtin_amdgcn_wmma_f32_16x16x4_f32(false, a1, false, band[c], (short)0, d1, false, false);
  }

  // ---- D-layout -> LDS (rows 0-15 from d0; rows 16-23 from d1 upper half) ----
  #pragma unroll
  for (int r = 0; r < 8; ++r)
    lm[(r + 8 * kh) * MS + n] = d0[r];
  #pragma unroll
  for (int r = 0; r < 8; ++r)
    if (kh) lm[(16 + r) * MS + n] = d1[r];
  __syncthreads();

  // ---- Vertical pass: Out(16x16) = R(16x24) * M(24x16), A = band ----
  v8f acc = {};
  #pragma unroll
  for (int c = 0; c < HALO / 4; ++c) {
    const int k0 = 4 * c + 2 * kh;
    v2f b;                                            // B rows k0,k0+1, col n
    b.x = lm[k0 * MS + n];
    b.y = lm[(k0 + 1) * MS + n];
    acc = __builtin_amdgcn_wmma_f32_16x16x4_f32(false, band[c], false, b, (short)0, acc, false, false);
  }

  // ---- Store 16x16 output tile (C/D layout -> row-major global) ----
  const long obase = base + (long)(r0 + 8 * kh) * IMW + (c0 + n);
  #pragma unroll
  for (int r = 0; r < 8; ++r)
    out[obase + (long)r * IMW] = acc[r];
}

extern "C" void kernel_launch(void* const* d_in, const int* in_sizes, int n_in,
                              void* d_out, int out_size, void* d_ws, size_t ws_size,
                              hipStream_t stream) {
  (void)n_in; (void)out_size; (void)d_ws; (void)ws_size;
  const float* x = (const float*)d_in[0];
  float* out = (float*)d_out;
  const int images = in_sizes[0] / (IMW * IMH);              // 24
  const int tiles  = images * (IMH / TILE) * (IMW / TILE);   // 98304
  const int blocks = tiles / WAVES;                          // exact division
  box9_wmma<<<blocks, WAVES * 32, 0, stream>>>(x, out);
}